// testerOrder2_55422257987994
// MI455X (gfx1250) — compile-verified
//
#include <hip/hip_runtime.h>

typedef __attribute__((ext_vector_type(2))) float v2f;
typedef __attribute__((ext_vector_type(8))) float v8f;

#define NVERT 48
#define KN    16           // neighbors (incl. self)
#define DD    2
#define CH    36           // channels = 18 contractions * D
#define CSTRIDE 40         // padded row stride for collapsed tile in LDS
#define WPAD  48           // padded output columns (3 x 16)

#if __has_builtin(__builtin_amdgcn_wmma_f32_16x16x4_f32)
#define HAVE_WMMA_F32_4 1
#endif

// One block per vertex. Builds the 256x36 "collapsed" activation tile entirely
// in LDS, then does the (256x36) x (36x36)^T GEMM with v_wmma_f32_16x16x4_f32,
// fuses bias+relu+fc_w weighting, and writes one fp32 partial per vertex.
__global__ __launch_bounds__(256) void tester55422_fused_kernel(
    const float* __restrict__ X, const int* __restrict__ nbr,
    const float* __restrict__ w1_w, const float* __restrict__ w1_b,
    const float* __restrict__ fc_w, float* __restrict__ partial)
{
    __shared__ int   s_nbr[KN];
    __shared__ int   s_nn[KN * KN];
    __shared__ float s_R[KN * KN];
    __shared__ float s_S[KN];
    __shared__ float s_Xv[KN][DD];
    __shared__ float s_A2[KN * KN][DD];
    __shared__ float s_B3[KN][DD];
    __shared__ float s_C1[DD], s_C2[DD];
    __shared__ float s_col[KN * KN * CSTRIDE];   // 256 rows x 40 = 40 KB
    __shared__ float s_W[CH * WPAD];             // W^T, padded cols
    __shared__ float s_bias[WPAD];
    __shared__ float s_fc[WPAD];
    __shared__ float s_red[8];

    const int i   = blockIdx.x;
    const int tid = threadIdx.x;
    const int xm  = tid >> 4;    // acts as m or x
    const int ya  = tid & 15;    // acts as a or y

    if (tid < KN) s_nbr[tid] = nbr[i * KN + tid];
    __syncthreads();

    // second-hop neighbor table: nn[m][q] = nbr[nbr[i][m]][q]
    s_nn[tid] = nbr[s_nbr[xm] * KN + ya];
    if (tid < KN * DD) {
        int m = tid >> 1, c = tid & 1;
        s_Xv[m][c] = X[s_nbr[m] * DD + c];
    }
    __syncthreads();

    // R[m][a] = 1 iff nbr[i][a] appears in nbr[nbr[i][m]]
    {
        int v = s_nbr[ya];
        float r = 0.f;
        #pragma unroll
        for (int q = 0; q < KN; ++q) r = (s_nn[xm * KN + q] == v) ? 1.f : r;
        s_R[tid] = r;
    }
    __syncthreads();

    if (tid < KN) {
        float s = 0.f;
        #pragma unroll
        for (int a = 0; a < KN; ++a) s += s_R[tid * KN + a];
        s_S[tid] = s;
    }
    __syncthreads();

    // A2[x][y][c] = sum_m Xv[m][c] R[m][x] R[m][y]
    {
        float a0 = 0.f, a1 = 0.f;
        #pragma unroll
        for (int m = 0; m < KN; ++m) {
            float rr = s_R[m * KN + xm] * s_R[m * KN + ya];
            a0 += s_Xv[m][0] * rr;
            a1 += s_Xv[m][1] * rr;
        }
        s_A2[tid][0] = a0; s_A2[tid][1] = a1;
    }
    if (tid < KN * DD) {
        int x = tid >> 1, c = tid & 1;
        float b = 0.f;
        #pragma unroll
        for (int m = 0; m < KN; ++m) b += s_Xv[m][c] * s_R[m * KN + x] * s_S[m];
        s_B3[x][c] = b;
    }
    if (tid < DD) {
        float c1 = 0.f, c2 = 0.f;
        #pragma unroll
        for (int m = 0; m < KN; ++m) {
            c1 += s_Xv[m][tid] * s_S[m] * s_S[m];
            c2 += s_Xv[m][tid];
        }
        s_C1[tid] = c1; s_C2[tid] = c2;
    }
    __syncthreads();

    // Build collapsed[x][y][0..35] (channel order matches jnp.concatenate)
    {
        float* row  = &s_col[tid * CSTRIDE];
        float r     = s_R[xm * KN + ya];
        float s     = s_S[xm];
        float diag  = (xm == ya) ? 1.f : 0.f;
        #pragma unroll
        for (int c = 0; c < DD; ++c) {
            float xv   = s_Xv[xm][c];
            float base = xv * r * s;          // t5
            float k16  = 16.f * base;         // t0 and the 9 c12_id copies
            row[0  + c] = k16;                            // t0
            row[2  + c] = xv * s * s;                     // t1
            row[4  + c] = 16.f * s_A2[tid][c];            // t2
            row[6  + c] = s_B3[xm][c];                    // t3
            row[8  + c] = diag * s_C1[c];                 // t4
            row[10 + c] = base;                           // t5
            #pragma unroll
            for (int t = 0; t < 9; ++t) row[12 + 2 * t + c] = k16;  // c12_id x9
            row[30 + c] = xv * r;                         // nmzzfzc
            row[32 + c] = s_Xv[ya][c] * s_R[ya * KN + xm];// nzazfzc
            row[34 + c] = diag * s_C2[c];                 // nzzzfgc
        }
        row[36] = row[37] = row[38] = row[39] = 0.f;
    }

    // Load W^T padded: s_W[v * WPAD + j] = w1_w[j][v], zero for j >= 36
    for (int idx = tid; idx < CH * WPAD; idx += 256) {
        int v = idx / WPAD, j = idx % WPAD;
        s_W[idx] = (j < CH) ? w1_w[j * CH + v] : 0.f;
    }
    if (tid < WPAD) {
        s_bias[tid] = (tid < CH) ? w1_b[tid] : 0.f;
        s_fc[tid]   = (tid < CH) ? fc_w[tid] : 0.f;
    }
    __syncthreads();

    // GEMM 256x36 @ 36x48(padded) via v_wmma_f32_16x16x4_f32,
    // fused bias + relu + fc weighting + reduction.
    const int wave  = tid >> 5;
    const int lane  = tid & 31;
    const int mn    = lane & 15;
    const int khalf = (lane >> 4) << 1;   // lanes 0-15 -> K+{0,1}; 16-31 -> K+{2,3}
    float wacc = 0.f;

    for (int task = wave; task < 48; task += 8) {   // uniform per wave: EXEC all-1s
        const int rt = task / 3, ct = task % 3;
        v8f acc = {0.f, 0.f, 0.f, 0.f, 0.f, 0.f, 0.f, 0.f};
#ifdef HAVE_WMMA_F32_4
        #pragma unroll
        for (int kk = 0; kk < 9; ++kk) {
            int k0 = kk * 4 + khalf;
            v2f a, b;
            a.x = s_col[(rt * 16 + mn) * CSTRIDE + k0];
            a.y = s_col[(rt * 16 + mn) * CSTRIDE + k0 + 1];
            b.x = s_W[k0 * WPAD + ct * 16 + mn];
            b.y = s_W[(k0 + 1) * WPAD + ct * 16 + mn];
            acc = __builtin_amdgcn_wmma_f32_16x16x4_f32(
                false, a, false, b, (short)0, acc, false, false);
        }
#else
        // Scalar fallback with identical C-fragment layout semantics
        #pragma unroll
        for (int r2 = 0; r2 < 8; ++r2) {
            int Mr = r2 + ((lane >> 4) << 3);
            float d = 0.f;
            for (int v = 0; v < CH; ++v)
                d += s_col[(rt * 16 + Mr) * CSTRIDE + v] * s_W[v * WPAD + ct * 16 + mn];
            acc[r2] = d;
        }
#endif
        float bias = s_bias[ct * 16 + mn];
        float fcw  = s_fc[ct * 16 + mn];
        #pragma unroll
        for (int r2 = 0; r2 < 8; ++r2) {
            float v = acc[r2] + bias;
            wacc += (v > 0.f ? v : 0.f) * fcw;   // padded cols contribute 0
        }
    }

    // wave32 reduction, then cross-wave via LDS (deterministic order)
    for (int off = 16; off > 0; off >>= 1) wacc += __shfl_xor(wacc, off, 32);
    if (lane == 0) s_red[wave] = wacc;
    __syncthreads();
    if (tid == 0) {
        float p = 0.f;
        #pragma unroll
        for (int w = 0; w < 8; ++w) p += s_red[w];
        partial[i] = p;
    }
}

__global__ void tester55422_reduce_kernel(const float* __restrict__ partial,
                                          const float* __restrict__ fc_b,
                                          float* __restrict__ out)
{
    if (threadIdx.x == 0) {
        float s = 0.f;
        #pragma unroll
        for (int i = 0; i < NVERT; ++i) s += partial[i];
        out[0] = s + fc_b[0];
    }
}

extern "C" void kernel_launch(void* const* d_in, const int* in_sizes, int n_in,
                              void* d_out, int out_size, void* d_ws, size_t ws_size,
                              hipStream_t stream) {
    const float* X    = (const float*)d_in[0];
    const int*   nbr  = (const int*)  d_in[1];
    const float* w1_w = (const float*)d_in[2];
    const float* w1_b = (const float*)d_in[3];
    const float* fc_w = (const float*)d_in[4];
    const float* fc_b = (const float*)d_in[5];
    float* out  = (float*)d_out;
    float* pws  = (float*)d_ws;   // 48 fp32 partials

    tester55422_fused_kernel<<<NVERT, 256, 0, stream>>>(X, nbr, w1_w, w1_b, fc_w, pws);
    tester55422_reduce_kernel<<<1, 32, 0, stream>>>(pws, fc_b, out);
}